// GraphModule_27195732918373
// MI455X (gfx1250) — compile-verified
//
#include <hip/hip_runtime.h>

static constexpr int NN = 2048;  // nodes
static constexpr int DD = 128;   // feature dim
static constexpr int RR = 43;    // relation types

typedef __attribute__((ext_vector_type(2))) float v2f;
typedef __attribute__((ext_vector_type(8))) float v8f;

// ---------------------------------------------------------------------------
// Kernel 0: head[i] = inputs[i,:]·w_head + b_head ; tail likewise. 2048 dots.
// ---------------------------------------------------------------------------
__global__ void head_tail_kernel(const float* __restrict__ x,
                                 const float* __restrict__ w_head,
                                 const float* __restrict__ b_head,
                                 const float* __restrict__ w_tail,
                                 const float* __restrict__ b_tail,
                                 float* __restrict__ head,
                                 float* __restrict__ tail) {
  int i = blockIdx.x * blockDim.x + threadIdx.x;
  if (i >= NN) return;
  const float* row = x + (size_t)i * DD;
  float h = 0.f, t = 0.f;
#pragma unroll 8
  for (int d = 0; d < DD; ++d) {
    float v = row[d];
    h = fmaf(v, w_head[d], h);
    t = fmaf(v, w_tail[d], t);
  }
  head[i] = h + b_head[0];
  tail[i] = t + b_tail[0];
}

// ---------------------------------------------------------------------------
// Kernel 1: logitsT[j,i] = rel_mask[i,j] + head[i] + tail[j] + b_rel
//                          + sum_r relation[i,j,r]*w_rel[r]
// Streams the 721 MB relation tensor with j varying across adjacent lanes
// (reads stay cache-friendly), transposes a 32x32 tile through LDS so the
// j-major store of the NxN logits is fully coalesced.
// ---------------------------------------------------------------------------
__global__ void __launch_bounds__(256) logits_kernel(
    const float* __restrict__ relation,
    const float* __restrict__ rel_mask,
    const float* __restrict__ w_rel,
    const float* __restrict__ b_rel,
    const float* __restrict__ head,
    const float* __restrict__ tail,
    float* __restrict__ logitsT) {
  __shared__ float s[32][33];   // +1 pad: conflict-free transposed read
  __shared__ float sw[RR];
  int t = threadIdx.x;
  if (t < RR) sw[t] = w_rel[t];
  __syncthreads();

  int i0 = blockIdx.y * 32;
  int j0 = blockIdx.x * 32;
  float br = b_rel[0];

#pragma unroll
  for (int q = 0; q < 4; ++q) {
    int p  = t + 256 * q;
    int ii = p >> 5, jj = p & 31;
    int i = i0 + ii, j = j0 + jj;
    const float* rel = relation + ((size_t)i * NN + j) * RR;
    if (q < 3) {  // hint next chunk of the big stream into cache
      int p2  = p + 256;
      int i2  = i0 + (p2 >> 5), j2 = j0 + (p2 & 31);
      __builtin_prefetch(relation + ((size_t)i2 * NN + j2) * RR, 0, 0);
    }
    float acc = br + head[i] + tail[j] + rel_mask[(size_t)i * NN + j];
#pragma unroll
    for (int r = 0; r < RR; ++r) acc = fmaf(rel[r], sw[r], acc);
    s[ii][jj] = acc;
  }
  __syncthreads();

#pragma unroll
  for (int q = 0; q < 4; ++q) {
    int p  = t + 256 * q;
    int jj = p >> 5, ii = p & 31;   // transposed assignment
    logitsT[(size_t)(j0 + jj) * NN + (i0 + ii)] = s[ii][jj];
  }
}

// ---------------------------------------------------------------------------
// Kernel 2: softmax over axis=0 of the original weight == row softmax of
// logitsT. One block per column j; 8 values/thread in registers; LDS tree
// reductions; normalize in place.
// ---------------------------------------------------------------------------
__global__ void __launch_bounds__(256) softmax_kernel(float* __restrict__ logitsT) {
  __shared__ float red[256];
  float* row = logitsT + (size_t)blockIdx.x * NN;
  int t = threadIdx.x;

  float v[8];
  float mx = -__builtin_inff();
#pragma unroll
  for (int q = 0; q < 8; ++q) {
    v[q] = row[t + 256 * q];
    mx = fmaxf(mx, v[q]);
  }
  red[t] = mx;
  __syncthreads();
  for (int s = 128; s > 0; s >>= 1) {
    if (t < s) red[t] = fmaxf(red[t], red[t + s]);
    __syncthreads();
  }
  mx = red[0];
  __syncthreads();

  float sum = 0.f;
#pragma unroll
  for (int q = 0; q < 8; ++q) {
    v[q] = __expf(v[q] - mx);
    sum += v[q];
  }
  red[t] = sum;
  __syncthreads();
  for (int s = 128; s > 0; s >>= 1) {
    if (t < s) red[t] += red[t + s];
    __syncthreads();
  }
  float inv = 1.0f / red[0];

#pragma unroll
  for (int q = 0; q < 8; ++q) row[t + 256 * q] = v[q] * inv;
}

// ---------------------------------------------------------------------------
// Kernel 3: out[i,d] = sum_j P[i,j] * x[j,d], with P stored transposed
// (pT[j,i]). One wave per 16x16 output tile, K-loop step 4 via
// V_WMMA_F32_16X16X4_F32 (fp32 A/B/C — exact precision, matrix pipe).
// A-frag layout (16x4 f32, ISA 7.12.2): lanes 0-15 hold K=0(v0)/K=1(v1),
// lanes 16-31 hold K=2(v0)/K=3(v1); M = lane%16. B symmetric with N = lane%16.
// C/D: VGPR v -> M=v (lanes 0-15) / M=v+8 (lanes 16-31), N = lane%16.
// ---------------------------------------------------------------------------
__global__ void __launch_bounds__(128) matmul_wmma_kernel(
    const float* __restrict__ pT,
    const float* __restrict__ x,
    float* __restrict__ out) {
  int lane = threadIdx.x & 31;
  int wv   = threadIdx.x >> 5;
  int i0 = (blockIdx.x * 4 + wv) * 16;   // output row tile
  int d0 = blockIdx.y * 16;              // output col tile
  int half = lane >> 4;                  // 0: K pair {0,1}, 1: K pair {2,3}
  int l    = lane & 15;

  v8f c = {};
  for (int k0 = 0; k0 < NN; k0 += 4) {
    int ka = k0 + 2 * half;
    v2f a, b;
    a.x = pT[(size_t)ka * NN + i0 + l];        // A[m=l, k=ka]   = P[i0+l, ka]
    a.y = pT[(size_t)(ka + 1) * NN + i0 + l];  // A[m=l, k=ka+1]
    b.x = x[(size_t)ka * DD + d0 + l];         // B[k=ka,   n=l]
    b.y = x[(size_t)(ka + 1) * DD + d0 + l];   // B[k=ka+1, n=l]
    c = __builtin_amdgcn_wmma_f32_16x16x4_f32(
        /*neg_a=*/false, a, /*neg_b=*/false, b,
        /*c_mod=*/(short)0, c, /*reuse_a=*/false, /*reuse_b=*/false);
  }

#pragma unroll
  for (int v = 0; v < 8; ++v)
    out[(size_t)(i0 + v + 8 * half) * DD + (d0 + l)] = c[v];
}

// ---------------------------------------------------------------------------
extern "C" void kernel_launch(void* const* d_in, const int* in_sizes, int n_in,
                              void* d_out, int out_size, void* d_ws, size_t ws_size,
                              hipStream_t stream) {
  (void)in_sizes; (void)n_in; (void)out_size; (void)ws_size;
  const float* inputs   = (const float*)d_in[0];   // [N, D]
  const float* relation = (const float*)d_in[1];   // [N, N, R]
  const float* rel_mask = (const float*)d_in[2];   // [N, N]
  const float* w_rel    = (const float*)d_in[3];   // [R, 1]
  const float* b_rel    = (const float*)d_in[4];   // [1]
  const float* w_head   = (const float*)d_in[5];   // [D, 1]
  const float* b_head   = (const float*)d_in[6];   // [1]
  const float* w_tail   = (const float*)d_in[7];   // [D, 1]
  const float* b_tail   = (const float*)d_in[8];   // [1]
  float* out = (float*)d_out;                      // [N, D]

  // Workspace layout: head[N] | tail[N] | logitsT[N*N]  (~16.8 MB)
  float* head    = (float*)d_ws;
  float* tail    = head + NN;
  float* logitsT = tail + NN;

  head_tail_kernel<<<NN / 256, 256, 0, stream>>>(inputs, w_head, b_head,
                                                 w_tail, b_tail, head, tail);

  logits_kernel<<<dim3(NN / 32, NN / 32), 256, 0, stream>>>(
      relation, rel_mask, w_rel, b_rel, head, tail, logitsT);

  softmax_kernel<<<NN, 256, 0, stream>>>(logitsT);

  matmul_wmma_kernel<<<dim3(NN / 16 / 4, DD / 16), 128, 0, stream>>>(
      logitsT, inputs, out);
}